// RGCN_CSR_Layer_69904887710374
// MI455X (gfx1250) — compile-verified
//
#include <hip/hip_runtime.h>
#include <math.h>

// RGCN CSR layer for MI455X (gfx1250, wave32).
//
// Structure: dense per-relation WMMA GEMM (f32 16x16x4) -> per-node CSR
// reduction (no atomics: each destination node owns its edge range), twice,
// with ReLU after layer 1 and a wave-level log_softmax after layer 2.
//
// Roofline: ~8 GFLOP total vs ~650 MB of traffic whose gather sources fit in
// the 192 MB L2 -> memory/latency bound, so keep f32 precision and use the
// f32 WMMA path; the GEMM stage exists mainly to feed the gathers from L2.

typedef __attribute__((ext_vector_type(2))) float v2f;
typedef __attribute__((ext_vector_type(8))) float v8f;

#define DIN 64  // inner dim of both layers (D_IN = D_H = 64)

// Y[n, r, 0:DOUT] = X[n, :] @ W[r, :, 0:DOUT]
// Block: 256 threads = 8 waves; covers 128 rows of X for one relation.
// Each wave computes a 16 x DOUTPAD tile with v_wmma_f32_16x16x4_f32.
template<int R, int DOUT, int DOUTPAD>
__global__ __launch_bounds__(256)
void rgcn_gemm_wmma(const float* __restrict__ X, const float* __restrict__ W,
                    float* __restrict__ Y, int N)
{
    constexpr int NT = DOUTPAD / 16;
    constexpr int RD = R * DOUT;             // Y row stride (elements)
    __shared__ float lA[128 * DIN];          // A tile, row-major
    __shared__ float lWt[DOUTPAD * DIN];     // W^T tile: lWt[n][k] = W[r][k][n]

    const int r       = blockIdx.y;
    const int rowBase = blockIdx.x * 128;
    const int tid     = threadIdx.x;

    // Cooperative, coalesced load of 128x64 A tile (zero-fill past N).
    const float* Xb = X + (size_t)rowBase * DIN;
    const bool fullBlock = (rowBase + 128 <= N);
    #pragma unroll
    for (int it = 0; it < (128 * DIN) / 256; ++it) {
        const int off = tid + it * 256;
        lA[off] = (fullBlock || (rowBase + (off >> 6) < N)) ? Xb[off] : 0.0f;
    }
    // Load W[r] transposed into LDS; zero-fill padded columns (n >= DOUT).
    const float* Wr = W + (size_t)r * DIN * DOUT;
    #pragma unroll
    for (int it = 0; it < (DOUTPAD * DIN) / 256; ++it) {
        const int off = tid + it * 256;
        const int n = off >> 6, k = off & 63;
        lWt[off] = (n < DOUT) ? Wr[k * DOUT + n] : 0.0f;
    }
    __syncthreads();

    const int wv   = tid >> 5;
    const int lane = tid & 31;
    const int half = lane >> 4;   // K split: lanes 0-15 take k,k+1; 16-31 take k+2,k+3
    const int l16  = lane & 15;

    // A operand base: row (wv*16 + l16), starting K offset half*2.
    const float* Arow = &lA[(wv * 16 + l16) * DIN + half * 2];
    // Uniform output base for this (block, relation).
    float* Yb = Y + (size_t)rowBase * RD + r * DOUT;

    #pragma unroll
    for (int t = 0; t < NT; ++t) {
        const int colBase = t * 16 + l16;
        const float* Bcol = &lWt[colBase * DIN + half * 2];
        v8f c = {};
        #pragma unroll
        for (int i = 0; i < DIN / 4; ++i) {   // 16 WMMA steps, K = 4 each
            v2f a = *(const v2f*)(Arow + 4 * i);
            v2f b = *(const v2f*)(Bcol + 4 * i);
            c = __builtin_amdgcn_wmma_f32_16x16x4_f32(
                    /*neg_a=*/false, a, /*neg_b=*/false, b,
                    /*c_mod=*/(short)0, c, /*reuse_a=*/false, /*reuse_b=*/false);
        }
        // C/D layout: VGPR j holds (M=j, N=lane) for lanes 0-15 and
        // (M=8+j, N=lane-16) for lanes 16-31.
        if (DOUT == DOUTPAD || colBase < DOUT) {
            // One per-lane address; the 8 rows differ by RD*4 bytes, which
            // fits global_store_b32's 24-bit immediate offset.
            float* p = Yb + (wv * 16 + half * 8) * RD + colBase;
            if (fullBlock) {
                #pragma unroll
                for (int j = 0; j < 8; ++j) p[j * RD] = c[j];
            } else {
                const int mBase = rowBase + wv * 16 + half * 8;
                #pragma unroll
                for (int j = 0; j < 8; ++j)
                    if (mBase + j < N) p[j * RD] = c[j];
            }
        }
    }
}

// H[i, :] = relu( sum_{e in [ptr[i],ptr[i+1])} XW[idx[e], rel[e], :] )
// One wave per destination node; 64 floats -> float2 per lane.
template<int R>
__global__ __launch_bounds__(256)
void rgcn_agg_relu(const float* __restrict__ XW, const int* __restrict__ ptr,
                   const int* __restrict__ idx, const int* __restrict__ rel,
                   float* __restrict__ H, int N)
{
    const int wv   = threadIdx.x >> 5;
    const int lane = threadIdx.x & 31;
    const int node = blockIdx.x * 8 + wv;
    if (node >= N) return;

    const int e0 = ptr[node], e1 = ptr[node + 1];   // uniform -> s_loads
    float ax = 0.0f, ay = 0.0f;
    for (int e = e0; e < e1; ++e) {
        const int src = idx[e];
        const int rv  = rel[e];
        const float2 v = *(const float2*)(XW + ((size_t)src * R + rv) * 64 + lane * 2);
        ax += v.x; ay += v.y;
    }
    float2 o; o.x = fmaxf(ax, 0.0f); o.y = fmaxf(ay, 0.0f);
    *(float2*)(H + (size_t)node * 64 + lane * 2) = o;
}

// out[i, :] = log_softmax( sum_e HW[idx[e], rel[e], 0:DOUT] ), DOUT = 40.
// One wave per node; lanes 0..DOUT/2-1 each hold 2 logits; wave-wide
// shuffle reductions for max and sum(exp).
template<int R, int DOUT>
__global__ __launch_bounds__(256)
void rgcn_agg_logsoftmax(const float* __restrict__ HW, const int* __restrict__ ptr,
                         const int* __restrict__ idx, const int* __restrict__ rel,
                         float* __restrict__ out, int N)
{
    const int wv   = threadIdx.x >> 5;
    const int lane = threadIdx.x & 31;
    const int node = blockIdx.x * 8 + wv;
    if (node >= N) return;

    const int  e0  = ptr[node], e1 = ptr[node + 1];
    const bool act = (lane * 2 < DOUT);
    float ax = 0.0f, ay = 0.0f;
    for (int e = e0; e < e1; ++e) {
        const int src = idx[e];
        const int rv  = rel[e];
        if (act) {
            const float2 v = *(const float2*)(HW + ((size_t)src * R + rv) * DOUT + lane * 2);
            ax += v.x; ay += v.y;
        }
    }
    float m = act ? fmaxf(ax, ay) : -__builtin_inff();
    #pragma unroll
    for (int off = 16; off >= 1; off >>= 1) m = fmaxf(m, __shfl_xor(m, off, 32));
    float s = act ? (__expf(ax - m) + __expf(ay - m)) : 0.0f;
    #pragma unroll
    for (int off = 16; off >= 1; off >>= 1) s += __shfl_xor(s, off, 32);
    const float ls = __logf(s);
    if (act) {
        float2 o; o.x = (ax - m) - ls; o.y = (ay - m) - ls;
        *(float2*)(out + (size_t)node * DOUT + lane * 2) = o;
    }
}

extern "C" void kernel_launch(void* const* d_in, const int* in_sizes, int n_in,
                              void* d_out, int out_size, void* d_ws, size_t ws_size,
                              hipStream_t stream)
{
    const float* x   = (const float*)d_in[0];
    const int*   ptr = (const int*)d_in[1];
    const int*   idx = (const int*)d_in[2];
    const int*   rel = (const int*)d_in[3];
    const float* W1  = (const float*)d_in[4];
    const float* W2  = (const float*)d_in[5];
    float* out = (float*)d_out;

    const int N = in_sizes[1] - 1;           // ptr has N+1 entries
    constexpr int R = 8, DH = 64, DOUT = 40;

    // Workspace: [ xw1: N*R*64 floats (reused as hw2: N*R*40) | h: N*64 ]
    // Peak: 75000*(512+64)*4 B = 172.8 MB.
    float* xw1 = (float*)d_ws;
    float* h   = xw1 + (size_t)N * R * DH;

    const int mblocks   = (N + 127) / 128;
    const int aggblocks = (N + 7) / 8;

    // Layer 1: xw1 = x @ W1 (per relation), then CSR-sum + ReLU -> h.
    rgcn_gemm_wmma<R, DH, DH><<<dim3(mblocks, R), 256, 0, stream>>>(x, W1, xw1, N);
    rgcn_agg_relu<R><<<aggblocks, 256, 0, stream>>>(xw1, ptr, idx, rel, h, N);
    // Layer 2: hw2 = h @ W2 (DOUT=40 padded to 48), reusing xw1's region,
    // then CSR-sum + log_softmax -> out.
    rgcn_gemm_wmma<R, DOUT, 48><<<dim3(mblocks, R), 256, 0, stream>>>(h, W2, xw1, N);
    rgcn_agg_logsoftmax<R, DOUT><<<aggblocks, 256, 0, stream>>>(xw1, ptr, idx, rel, out, N);
}